// GCN_75265006895335
// MI455X (gfx1250) — compile-verified
//
#include <hip/hip_runtime.h>
#include <hip/hip_bf16.h>
#include <math.h>

#define N_NODES 50000
#define N_EDGES 600000
#define IN_DIM  128
#define HID_DIM 128
#define OUT_DIM 40
#define BN_EPS  1e-5f

typedef __attribute__((ext_vector_type(2))) float v2f;
typedef __attribute__((ext_vector_type(4))) float v4f;
typedef __attribute__((ext_vector_type(8))) float v8f;

// ---------------------------------------------------------------- degree ----
__global__ void deg_init(float* deg, int n) {
    int i = blockIdx.x * blockDim.x + threadIdx.x;
    if (i < n) deg[i] = 1.0f;  // self-loop contributes 1
}

__global__ void deg_accum(const int* __restrict__ dst, float* deg, int e) {
    int i = blockIdx.x * blockDim.x + threadIdx.x;
    if (i < e) atomicAdd(&deg[dst[i]], 1.0f);
}

__global__ void deg_rsqrt(float* deg, int n) {
    int i = blockIdx.x * blockDim.x + threadIdx.x;
    if (i < n) deg[i] = rsqrtf(deg[i]);  // deg >= 1 always (self-loops)
}

// ------------------------------------------------------- GEMM: [n,128]x[128,128]
// One wave -> 16 rows x 128 cols.  W staged transposed in LDS (Wt[n][k]) so
// B fragments are contiguous ds_load_b64.  fp32 WMMA 16x16x4.
__global__ __launch_bounds__(256) void gemm_n128(const float* __restrict__ X,
                                                 const float* __restrict__ W,
                                                 float* __restrict__ Y, int nrows) {
    __shared__ float Wt[128 * 128];
    const int tid = threadIdx.x;
    for (int i = tid * 4; i < 128 * 128; i += 256 * 4) {
        v4f w = *(const v4f*)(W + i);          // W row-major [k][n]
        int k = i >> 7, n = i & 127;
        Wt[(n + 0) * 128 + k] = w.x;
        Wt[(n + 1) * 128 + k] = w.y;
        Wt[(n + 2) * 128 + k] = w.z;
        Wt[(n + 3) * 128 + k] = w.w;
    }
    __syncthreads();

    const int wave = tid >> 5, lane = tid & 31;
    const int l16 = lane & 15, lh = lane >> 4;   // lanes 16-31 hold K+2,K+3
    const int r0 = blockIdx.x * 128 + wave * 16;
    if (r0 >= nrows) return;                     // wave-uniform (strips are full)

    const float* xrow = X + (r0 + l16) * 128 + 2 * lh;
    v8f acc[8] = {};
    for (int k = 0; k < 128; k += 4) {
        v2f a = *(const v2f*)(xrow + k);         // A: row=M(lane), 2 consecutive K
#pragma unroll
        for (int t = 0; t < 8; ++t) {
            v2f b = *(const v2f*)&Wt[(t * 16 + l16) * 128 + k + 2 * lh];
            acc[t] = __builtin_amdgcn_wmma_f32_16x16x4_f32(
                false, a, false, b, (short)0, acc[t], false, false);
        }
    }
#pragma unroll
    for (int t = 0; t < 8; ++t) {
        float* yb = Y + (r0 + 8 * lh) * 128 + t * 16 + l16;   // D layout: VGPR r -> M=r(+8)
#pragma unroll
        for (int r = 0; r < 8; ++r) yb[r * 128] = acc[t][r];
    }
}

// ------------------------------------------------------- GEMM: [n,128]x[128,40]
__global__ __launch_bounds__(256) void gemm_n40(const float* __restrict__ X,
                                                const float* __restrict__ W,
                                                float* __restrict__ Y, int nrows) {
    __shared__ float Wt[48 * 128];               // cols padded 40 -> 48
    const int tid = threadIdx.x;
    for (int i = tid; i < 48 * 128; i += 256) Wt[i] = 0.0f;
    __syncthreads();
    for (int i = tid; i < 128 * OUT_DIM; i += 256) {
        int k = i / OUT_DIM, n = i - k * OUT_DIM;
        Wt[n * 128 + k] = W[i];
    }
    __syncthreads();

    const int wave = tid >> 5, lane = tid & 31;
    const int l16 = lane & 15, lh = lane >> 4;
    const int r0 = blockIdx.x * 128 + wave * 16;
    if (r0 >= nrows) return;

    const float* xrow = X + (r0 + l16) * 128 + 2 * lh;
    v8f acc[3] = {};
    for (int k = 0; k < 128; k += 4) {
        v2f a = *(const v2f*)(xrow + k);
#pragma unroll
        for (int t = 0; t < 3; ++t) {
            v2f b = *(const v2f*)&Wt[(t * 16 + l16) * 128 + k + 2 * lh];
            acc[t] = __builtin_amdgcn_wmma_f32_16x16x4_f32(
                false, a, false, b, (short)0, acc[t], false, false);
        }
    }
#pragma unroll
    for (int t = 0; t < 3; ++t) {
        int col = t * 16 + l16;
        if (col < OUT_DIM) {
#pragma unroll
            for (int r = 0; r < 8; ++r)
                Y[(r0 + r + 8 * lh) * OUT_DIM + col] = acc[t][r];
        }
    }
}

// -------------------------------------------------- aggregation (gather+scatter)
// agg[i] = h[i] * dinv[i]^2   (self-loop term folded in; also zero-initializes)
__global__ void agg_init(const float* __restrict__ h, const float* __restrict__ dinv,
                         float* __restrict__ agg, int n, int d4) {
    int gid = blockIdx.x * blockDim.x + threadIdx.x;
    if (gid >= n * d4) return;
    int row = gid / d4;
    float di = dinv[row];
    float w = di * di;
    v4f v = *(const v4f*)(h + gid * 4);
    v4f o = {v.x * w, v.y * w, v.z * w, v.w * w};
    *(v4f*)(agg + gid * 4) = o;
}

// one wave per edge (D=128): 32 lanes x float4 chunk, coalesced gather, atomic scatter
__global__ void edge_scatter128(const int* __restrict__ srcI, const int* __restrict__ dstI,
                                const float* __restrict__ dinv,
                                const float* __restrict__ h, float* __restrict__ agg,
                                int e) {
    int gid = blockIdx.x * blockDim.x + threadIdx.x;
    int ed = gid >> 5;
    if (ed >= e) return;
    int c = (gid & 31) * 4;
    int s = srcI[ed], d = dstI[ed];
    float w = dinv[s] * dinv[d];
    v4f m = *(const v4f*)(h + s * 128 + c);
    float* ap = agg + d * 128 + c;
    atomicAdd(ap + 0, m.x * w);
    atomicAdd(ap + 1, m.y * w);
    atomicAdd(ap + 2, m.z * w);
    atomicAdd(ap + 3, m.w * w);
}

// 10 float4 chunks per edge (D=40)
__global__ void edge_scatter40(const int* __restrict__ srcI, const int* __restrict__ dstI,
                               const float* __restrict__ dinv,
                               const float* __restrict__ h, float* __restrict__ agg,
                               int e) {
    int gid = blockIdx.x * blockDim.x + threadIdx.x;
    int ed = gid / 10;
    if (ed >= e) return;
    int c = (gid - ed * 10) * 4;
    int s = srcI[ed], d = dstI[ed];
    float w = dinv[s] * dinv[d];
    v4f m = *(const v4f*)(h + s * OUT_DIM + c);
    float* ap = agg + d * OUT_DIM + c;
    atomicAdd(ap + 0, m.x * w);
    atomicAdd(ap + 1, m.y * w);
    atomicAdd(ap + 2, m.z * w);
    atomicAdd(ap + 3, m.w * w);
}

// ---------------------------------------------------------------- batchnorm ----
__global__ void zero_stats(float* stats) {
    if (threadIdx.x < 256) stats[threadIdx.x] = 0.0f;
}

__global__ __launch_bounds__(128) void bn_stats(const float* __restrict__ x,
                                                float* __restrict__ stats, int n) {
    int t = threadIdx.x;                 // column 0..127
    int r0 = blockIdx.x * 512;
    int r1 = r0 + 512; if (r1 > n) r1 = n;
    float s = 0.0f, ss = 0.0f;
    for (int r = r0; r < r1; ++r) {
        float v = x[r * 128 + t];
        s += v; ss += v * v;
    }
    atomicAdd(&stats[t], s);
    atomicAdd(&stats[128 + t], ss);
}

__global__ void bn_apply_relu(const float* __restrict__ agg, const float* __restrict__ stats,
                              const float* __restrict__ gamma, const float* __restrict__ beta,
                              float* __restrict__ out, int n) {
    int gid = blockIdx.x * blockDim.x + threadIdx.x;
    if (gid >= n * 32) return;
    int col = (gid & 31) * 4;
    v4f v = *(const v4f*)(agg + gid * 4);
    v4f s = *(const v4f*)(stats + col);
    v4f q = *(const v4f*)(stats + 128 + col);
    v4f g = *(const v4f*)(gamma + col);
    v4f b = *(const v4f*)(beta + col);
    float inv = 1.0f / (float)n;
    v4f o;
    {
        float m = s.x * inv, var = q.x * inv - m * m;
        o.x = fmaxf((v.x - m) * rsqrtf(var + BN_EPS) * g.x + b.x, 0.0f);
    }
    {
        float m = s.y * inv, var = q.y * inv - m * m;
        o.y = fmaxf((v.y - m) * rsqrtf(var + BN_EPS) * g.y + b.y, 0.0f);
    }
    {
        float m = s.z * inv, var = q.z * inv - m * m;
        o.z = fmaxf((v.z - m) * rsqrtf(var + BN_EPS) * g.z + b.z, 0.0f);
    }
    {
        float m = s.w * inv, var = q.w * inv - m * m;
        o.w = fmaxf((v.w - m) * rsqrtf(var + BN_EPS) * g.w + b.w, 0.0f);
    }
    *(v4f*)(out + gid * 4) = o;
}

// --------------------------------------------------- bias + log_softmax (D=40)
__global__ __launch_bounds__(256) void bias_logsoftmax(const float* __restrict__ agg,
                                                       const float* __restrict__ bias,
                                                       float* __restrict__ out, int n) {
    int row = blockIdx.x * 8 + (threadIdx.x >> 5);
    int lane = threadIdx.x & 31;
    if (row >= n) return;
    const float* xr = agg + row * OUT_DIM;
    float x0 = xr[lane] + bias[lane];
    float x1 = (lane < 8) ? (xr[32 + lane] + bias[32 + lane]) : -INFINITY;
    float m = fmaxf(x0, x1);
    for (int off = 16; off; off >>= 1) m = fmaxf(m, __shfl_xor(m, off, 32));
    float e0 = expf(x0 - m);
    float e1 = (lane < 8) ? expf(x1 - m) : 0.0f;
    float sum = e0 + e1;
    for (int off = 16; off; off >>= 1) sum += __shfl_xor(sum, off, 32);
    float lse = m + logf(sum);
    out[row * OUT_DIM + lane] = x0 - lse;
    if (lane < 8) out[row * OUT_DIM + 32 + lane] = x1 - lse;
}

// ------------------------------------------------------------------- driver ----
extern "C" void kernel_launch(void* const* d_in, const int* in_sizes, int n_in,
                              void* d_out, int out_size, void* d_ws, size_t ws_size,
                              hipStream_t stream) {
    (void)in_sizes; (void)n_in; (void)out_size; (void)ws_size;
    const float* x   = (const float*)d_in[0];
    const int*   ei  = (const int*)d_in[1];
    const float* W1  = (const float*)d_in[2];
    const float* W2  = (const float*)d_in[4];
    const float* W3  = (const float*)d_in[6];
    const float* b3  = (const float*)d_in[7];
    const float* g1  = (const float*)d_in[8];
    const float* be1 = (const float*)d_in[9];
    const float* g2  = (const float*)d_in[10];
    const float* be2 = (const float*)d_in[11];
    float* out = (float*)d_out;

    char* ws = (char*)d_ws;
    const size_t SZ = (size_t)N_NODES * HID_DIM * sizeof(float);  // 25.6 MB
    float* bufA  = (float*)(ws);
    float* bufB  = (float*)(ws + SZ);
    float* dinv  = (float*)(ws + 2 * SZ);
    float* stats = (float*)(ws + 2 * SZ + (size_t)N_NODES * sizeof(float));

    const int* srcI = ei;
    const int* dstI = ei + N_EDGES;

    const int N = N_NODES, E = N_EDGES;
    dim3 blk(256);

    // degree / dinv
    deg_init <<<(N + 255) / 256, blk, 0, stream>>>(dinv, N);
    deg_accum<<<(E + 255) / 256, blk, 0, stream>>>(dstI, dinv, E);
    deg_rsqrt<<<(N + 255) / 256, blk, 0, stream>>>(dinv, N);

    // layer 1
    gemm_n128<<<(N + 127) / 128, blk, 0, stream>>>(x, W1, bufA, N);
    agg_init <<<(N * 32 + 255) / 256, blk, 0, stream>>>(bufA, dinv, bufB, N, 32);
    edge_scatter128<<<(E * 32 + 255) / 256, blk, 0, stream>>>(srcI, dstI, dinv, bufA, bufB, E);
    zero_stats<<<1, blk, 0, stream>>>(stats);
    bn_stats  <<<(N + 511) / 512, dim3(128), 0, stream>>>(bufB, stats, N);
    bn_apply_relu<<<(N * 32 + 255) / 256, blk, 0, stream>>>(bufB, stats, g1, be1, bufA, N);

    // layer 2
    gemm_n128<<<(N + 127) / 128, blk, 0, stream>>>(bufA, W2, bufB, N);
    agg_init <<<(N * 32 + 255) / 256, blk, 0, stream>>>(bufB, dinv, bufA, N, 32);
    edge_scatter128<<<(E * 32 + 255) / 256, blk, 0, stream>>>(srcI, dstI, dinv, bufB, bufA, E);
    zero_stats<<<1, blk, 0, stream>>>(stats);
    bn_stats  <<<(N + 511) / 512, dim3(128), 0, stream>>>(bufA, stats, N);
    bn_apply_relu<<<(N * 32 + 255) / 256, blk, 0, stream>>>(bufA, stats, g2, be2, bufB, N);

    // layer 3 + log_softmax
    gemm_n40 <<<(N + 127) / 128, blk, 0, stream>>>(bufB, W3, bufA, N);
    agg_init <<<(N * 10 + 255) / 256, blk, 0, stream>>>(bufA, dinv, bufB, N, 10);
    edge_scatter40<<<(E * 10 + 255) / 256, blk, 0, stream>>>(srcI, dstI, dinv, bufA, bufB, E);
    bias_logsoftmax<<<(N + 7) / 8, blk, 0, stream>>>(bufB, b3, out, N);
}